// countingLayer_77025943486830
// MI455X (gfx1250) — compile-verified
//
#include <hip/hip_runtime.h>
#include <stdint.h>

// Problem geometry (fixed by reference: x,u = (8,256,256,1) fp32, h = (8,1))
#define BATCH 8
#define HH    256
#define WW    256
#define NPIX  (HH * WW)          // 65536 pixels per image
#define SW    260                // padded LDS row stride (floats), 16B-aligned rows
#define PROWS 258                // padded rows (1 halo row top/bottom)
#define LDS_FLOATS (SW * PROWS)  // 67080 floats = 268320 B  (< 320KB WGP LDS)
#define EPSV  1e-5f

typedef unsigned int v4u __attribute__((ext_vector_type(4)));
typedef int          v8i __attribute__((ext_vector_type(8)));
typedef int          v4i __attribute__((ext_vector_type(4)));

// ---------------------------------------------------------------------------
// TDM: DMA one 256x256 fp32 image from global into the padded LDS interior.
// D# built per CDNA5 ISA. pad_enable inserts 4 dwords of
// skip after every 256 dwords, landing each row at padded (row+1, col 1..256)
// while hopping over the -inf halo column pair. clang-23 6-arg builtin:
// (u32x4 g0, i32x8 g1, i32x4 g2, i32x4 g3, i32x8 extra, i32 cpol).
// ---------------------------------------------------------------------------
__device__ __forceinline__ void tdm_load_image(uint32_t lds_byte_off, const float* gsrc) {
  uint64_t ga = (uint64_t)(uintptr_t)gsrc;
  v4u g0;
  g0.x = 1u;                                   // count=1, is_restore=0, gather=0
  g0.y = lds_byte_off;                         // lds_addr (bytes)
  g0.z = (uint32_t)(ga & 0xFFFFFFFFu);         // global_addr[31:0]
  g0.w = (uint32_t)((ga >> 32) & 0x01FFFFFFu)  // global_addr[56:32]
       | (2u << 30);                           // type = 2 ("image")
  v8i g1;
  g1[0] = (int)((2u << 16)      // data_size = 4B
              | (1u << 20)      // pad_enable
              | (7u << 22)      // pad_interval code 7 -> 256 dwords
              | (3u << 25));    // pad_amount  code 3 -> 4 dwords
  g1[1] = (int)(256u << 16);    // tensor_dim0 = 256 (lo16); atomic_barrier_addr=0
  g1[2] = (int)(256u << 16);    // tensor_dim0 hi16=0 | tensor_dim1 = 256 (lo16)
  g1[3] = (int)(256u << 16);    // tensor_dim1 hi16=0 | tile_dim0 = 256
  g1[4] = (int)256;             // tile_dim1 = 256 | tile_dim2 = 0 (2D)
  g1[5] = (int)256;             // tensor_dim0_stride = 256 (lo32)
  g1[6] = 0;                    // stride0 hi16 = 0 | tensor_dim1_stride lo16 = 0
  g1[7] = 1;                    // tensor_dim1_stride = 65536 (unused for 2D tile)
  v4i g2 = {0, 0, 0, 0};
  v4i g3 = {0, 0, 0, 0};
  v8i g4 = {0, 0, 0, 0, 0, 0, 0, 0};
  __builtin_amdgcn_tensor_load_to_lds(g0, g1, g2, g3, g4, 0);
}

// ---------------------------------------------------------------------------
// One chaotic (in-place) sweep of rec = min(maxpool3x3(rec), mask).
// Monotone non-decreasing, bounded by mask -> converges to the same least
// fixed point as the reference's Jacobi while_loop. Returns "changed".
// Each thread owns 16 aligned float4 chunks; 6 ds_load (b128+b64 per row),
// one global b128 mask read (L2-resident) per chunk.
// ---------------------------------------------------------------------------
__device__ __forceinline__ int gd_pass(float* __restrict__ P,
                                       const float* __restrict__ mask) {
  int changed = 0;
  for (int chunk = threadIdx.x; chunk < (NPIX / 4); chunk += blockDim.x) {
    const int p0 = chunk << 2;
    const int r  = p0 >> 8;      // original row
    const int c  = p0 & 255;     // original col (multiple of 4)
    const float* q0 = P + r * SW + c;        // padded row r   : cols c..c+5
    const float* q1 = q0 + SW;               // padded row r+1 (center row)
    const float* q2 = q1 + SW;               // padded row r+2
    float t0[6], t1[6], t2[6], hm[6];
#pragma unroll
    for (int j = 0; j < 6; ++j) { t0[j] = q0[j]; t1[j] = q1[j]; t2[j] = q2[j]; }
#pragma unroll
    for (int j = 0; j < 6; ++j) hm[j] = fmaxf(fmaxf(t0[j], t1[j]), t2[j]);

    const float4 m = *(const float4*)(mask + p0);
    float n0 = fminf(fmaxf(fmaxf(hm[0], hm[1]), hm[2]), m.x);
    float n1 = fminf(fmaxf(fmaxf(hm[1], hm[2]), hm[3]), m.y);
    float n2 = fminf(fmaxf(fmaxf(hm[2], hm[3]), hm[4]), m.z);
    float n3 = fminf(fmaxf(fmaxf(hm[3], hm[4]), hm[5]), m.w);

    float* ctr = (float*)(q1 + 1);
    if (n0 > t1[1]) { ctr[0] = n0; changed = 1; }
    if (n1 > t1[2]) { ctr[1] = n1; changed = 1; }
    if (n2 > t1[3]) { ctr[2] = n2; changed = 1; }
    if (n3 > t1[4]) { ctr[3] = n3; changed = 1; }
  }
  return changed;
}

// Block-wide fixed-point driver. Triple barrier: reset / set / read windows
// never overlap, and the terminating sweep performs zero LDS writes, so the
// exit decision is uniform and the state is a true fixed point.
__device__ void geodesic(float* P, const float* mask, int* s_flag) {
  for (;;) {
    if (threadIdx.x == 0) *s_flag = 0;
    __syncthreads();
    int ch = gd_pass(P, mask);
    if (ch) *s_flag = 1;
    __syncthreads();
    int f = *s_flag;
    __syncthreads();
    if (!f) break;
  }
}

__device__ __forceinline__ void init_borders(float* P) {
  const float NI = -__builtin_inff();
  for (int i = threadIdx.x; i < PROWS; i += blockDim.x) {
    P[i] = NI;                       // padded row 0, cols 0..257
    P[(PROWS - 1) * SW + i] = NI;    // padded row 257
  }
  for (int rr = threadIdx.x; rr < HH; rr += blockDim.x) {
    P[(rr + 1) * SW + 0]   = NI;     // left halo col
    P[(rr + 1) * SW + 257] = NI;     // right halo col
  }
}

// ---------------------------------------------------------------------------
// Main kernel: one block = one image, whole pipeline resident in 320KB LDS.
// ---------------------------------------------------------------------------
__global__ void __launch_bounds__(1024, 1)
hmaxima_kernel(const float* __restrict__ x, const float* __restrict__ h,
               const float* __restrict__ u,
               float* __restrict__ xh_ws, float* __restrict__ rmax_ws,
               float* __restrict__ cc_ws, int* __restrict__ one_ws,
               int* __restrict__ zero_ws) {
  extern __shared__ float P[];   // 260x258 padded fp32 image
  __shared__ int   s_flag;
  __shared__ int   s_one, s_zero;
  __shared__ float s_cc;

  const int b   = blockIdx.x;
  const int tid = threadIdx.x;
  const float* xb  = x + (size_t)b * NPIX;
  const float* ub  = u + (size_t)b * NPIX;
  float*       xhb = xh_ws   + (size_t)b * NPIX;
  float*       rmb = rmax_ws + (size_t)b * NPIX;
  const float  hb  = h[b];

  init_borders(P);
  __syncthreads();

  // --- Phase A: xh = GD(x - h, x) ---------------------------------------
  if (tid < 32) {  // one wave issues the TDM DMA (EXEC-independent, wave-level)
    tdm_load_image((uint32_t)(uintptr_t)(P + SW + 1), xb);
    __builtin_amdgcn_s_wait_tensorcnt(0);
  }
  __syncthreads();
  for (int p = tid; p < NPIX; p += blockDim.x) {
    const int idx = ((p >> 8) + 1) * SW + (p & 255) + 1;
    P[idx] -= hb;                          // marker = x - h (mask = x, global)
  }
  __syncthreads();
  geodesic(P, xb, &s_flag);                // P = xh (padded)

  for (int p = tid; p < NPIX; p += blockDim.x) {
    const int idx = ((p >> 8) + 1) * SW + (p & 255) + 1;
    xhb[p] = P[idx];
  }
  __threadfence_block();
  __syncthreads();

  // --- Phase B: Rmax = (xh - GD(xh - EPS, xh) > 0) ----------------------
  for (int p = tid; p < NPIX; p += blockDim.x) {
    const int idx = ((p >> 8) + 1) * SW + (p & 255) + 1;
    P[idx] -= EPSV;
  }
  __syncthreads();
  geodesic(P, xhb, &s_flag);

  int any1 = 0, any0 = 0;
  for (int p = tid; p < NPIX; p += blockDim.x) {
    const int idx = ((p >> 8) + 1) * SW + (p & 255) + 1;
    const float rm = (xhb[p] - P[idx] > 0.0f) ? 1.0f : 0.0f;
    rmb[p] = rm;
    if (rm > 0.5f) any1 = 1; else any0 = 1;
  }
  if (tid == 0) { s_one = 0; s_zero = 0; s_cc = 0.0f; }
  __threadfence_block();
  __syncthreads();
  if (any1) atomicOr(&s_one, 1);
  if (any0) atomicOr(&s_zero, 1);

  // --- Phase C: R = GD(min(u/100, Rmax), Rmax); Detection = (U == R) ----
  for (int p = tid; p < NPIX; p += blockDim.x) {
    const int idx = ((p >> 8) + 1) * SW + (p & 255) + 1;
    P[idx] = fminf(ub[p] / 100.0f, rmb[p]);
  }
  __syncthreads();
  geodesic(P, rmb, &s_flag);

  float cc = 0.0f;
  for (int p = tid; p < NPIX; p += blockDim.x) {
    const int idx = ((p >> 8) + 1) * SW + (p & 255) + 1;
    cc += (ub[p] / 100.0f == P[idx]) ? 1.0f : 0.0f;
  }
  atomicAdd(&s_cc, cc);   // exact: integer-valued fp32 partials
  __syncthreads();
  if (tid == 0) { cc_ws[b] = s_cc; one_ws[b] = s_one; zero_ws[b] = s_zero; }
}

// CC_ = min(CC, LAMDA * (max(Rmax) - min(Rmax)) * CC), max/min global over batch.
__global__ void finalize_kernel(const float* __restrict__ cc_ws,
                                const int* __restrict__ one_ws,
                                const int* __restrict__ zero_ws,
                                float* __restrict__ out) {
  const int t = threadIdx.x;
  if (t < BATCH) {
    int anyOne = 0, anyZero = 0;
#pragma unroll
    for (int i = 0; i < BATCH; ++i) { anyOne |= one_ws[i]; anyZero |= zero_ws[i]; }
    const float gmax = anyOne ? 1.0f : 0.0f;
    const float gmin = anyZero ? 0.0f : 1.0f;
    const float cc = cc_ws[t];
    out[t] = fminf(cc, 100.0f * (gmax - gmin) * cc);
  }
}

extern "C" void kernel_launch(void* const* d_in, const int* in_sizes, int n_in,
                              void* d_out, int out_size, void* d_ws, size_t ws_size,
                              hipStream_t stream) {
  (void)in_sizes; (void)n_in; (void)out_size; (void)ws_size;
  const float* x = (const float*)d_in[0];
  const float* h = (const float*)d_in[1];
  const float* u = (const float*)d_in[2];
  float* out = (float*)d_out;

  // Workspace: xh (2MB) | rmax (2MB) | cc[8] | one[8] | zero[8]
  char* ws = (char*)d_ws;
  float* xh_ws   = (float*)(ws);
  float* rmax_ws = (float*)(ws + (size_t)BATCH * NPIX * 4);
  float* cc_ws   = (float*)(ws + (size_t)2 * BATCH * NPIX * 4);
  int*   one_ws  = (int*)  (ws + (size_t)2 * BATCH * NPIX * 4 + 64);
  int*   zero_ws = (int*)  (ws + (size_t)2 * BATCH * NPIX * 4 + 128);

  const size_t lds_bytes = (size_t)LDS_FLOATS * sizeof(float);  // 268320 B
  hmaxima_kernel<<<dim3(BATCH), dim3(1024), lds_bytes, stream>>>(
      x, h, u, xh_ws, rmax_ws, cc_ws, one_ws, zero_ws);
  finalize_kernel<<<dim3(1), dim3(32), 0, stream>>>(cc_ws, one_ws, zero_ws, out);
}